// Operator_66752381714490
// MI455X (gfx1250) — compile-verified
//
#include <hip/hip_runtime.h>

typedef __attribute__((ext_vector_type(2))) float v2f;
typedef __attribute__((ext_vector_type(8))) float v8f;

#define TPB 256
#define EPT 2   // elements per thread

// Per-element Tri3 energy: 0.5*|u_grad|^2 + 0.5*|u|^2, integrated over 3 quad points.
__device__ __forceinline__ float tri3_energy(
    const float* __restrict__ nod, const float* __restrict__ xy,
    const int* __restrict__ elems,
    const float* __restrict__ Nm, const float* __restrict__ dNm,
    const float* __restrict__ qw, int e)
{
    const int n0 = elems[3 * e + 0];
    const int n1 = elems[3 * e + 1];
    const int n2 = elems[3 * e + 2];

    const float2* __restrict__ xy2 = (const float2*)xy;
    const float2* __restrict__ nv2 = (const float2*)nod;

    const float2 c0 = xy2[n0], c1 = xy2[n1], c2 = xy2[n2];
    const float2 v0 = nv2[n0], v1 = nv2[n1], v2 = nv2[n2];

    float acc = 0.0f;
#pragma unroll
    for (int q = 0; q < 3; ++q) {
        const float dn00 = dNm[(q * 3 + 0) * 2 + 0], dn01 = dNm[(q * 3 + 0) * 2 + 1];
        const float dn10 = dNm[(q * 3 + 1) * 2 + 0], dn11 = dNm[(q * 3 + 1) * 2 + 1];
        const float dn20 = dNm[(q * 3 + 2) * 2 + 0], dn21 = dNm[(q * 3 + 2) * 2 + 1];
        const float N0 = Nm[q * 3 + 0], N1 = Nm[q * 3 + 1], N2 = Nm[q * 3 + 2];

        // J[d][j] = sum_a xy[a][d] * dN[q][a][j]
        const float J00 = c0.x * dn00 + c1.x * dn10 + c2.x * dn20;
        const float J01 = c0.x * dn01 + c1.x * dn11 + c2.x * dn21;
        const float J10 = c0.y * dn00 + c1.y * dn10 + c2.y * dn20;
        const float J11 = c0.y * dn01 + c1.y * dn11 + c2.y * dn21;

        const float det  = J00 * J11 - J01 * J10;
        const float rdet = 1.0f / det;
        const float i00 =  J11 * rdet, i01 = -J01 * rdet;
        const float i10 = -J10 * rdet, i11 =  J00 * rdet;

        // dNp[a][d] = sum_j dN[q][a][j] * invJ[j][d]
        const float p00 = dn00 * i00 + dn01 * i10, p01 = dn00 * i01 + dn01 * i11;
        const float p10 = dn10 * i00 + dn11 * i10, p11 = dn10 * i01 + dn11 * i11;
        const float p20 = dn20 * i00 + dn21 * i10, p21 = dn20 * i01 + dn21 * i11;

        // u_grad[v][d] = sum_a vals[a][v] * dNp[a][d]
        const float g00 = v0.x * p00 + v1.x * p10 + v2.x * p20;
        const float g01 = v0.x * p01 + v1.x * p11 + v2.x * p21;
        const float g10 = v0.y * p00 + v1.y * p10 + v2.y * p20;
        const float g11 = v0.y * p01 + v1.y * p11 + v2.y * p21;

        // u[v] = sum_a N[q][a] * vals[a][v]
        const float u0 = N0 * v0.x + N1 * v1.x + N2 * v2.x;
        const float u1 = N0 * v0.y + N1 * v1.y + N2 * v2.y;

        const float en = 0.5f * (g00 * g00 + g01 * g01 + g10 * g10 + g11 * g11)
                       + 0.5f * (u0 * u0 + u1 * u1);
        acc += en * det * qw[q];
    }
    return acc;
}

__global__ void __launch_bounds__(TPB) fea_energy_partials(
    const float* __restrict__ nod, const float* __restrict__ xy,
    const int* __restrict__ elems, const float* __restrict__ Nm,
    const float* __restrict__ dNm, const float* __restrict__ qw,
    int nElem, float* __restrict__ blockSums)
{
    const int tid = threadIdx.x;
    const int e0  = blockIdx.x * (TPB * EPT) + tid;
    const int e1  = e0 + TPB;

    float p0 = 0.0f, p1 = 0.0f;
    if (e0 < nElem) p0 = tri3_energy(nod, xy, elems, Nm, dNm, qw, e0);
    if (e1 < nElem) p1 = tri3_energy(nod, xy, elems, Nm, dNm, qw, e1);

    // --- Wave-level reduction via V_WMMA_F32_16X16X4_F32 ---
    // A (16x4 f32): lane L<16 supplies A[L][0]=p0, A[L][1]=p1;
    //               lane 16+L supplies A[L][2]=p0, A[L][3]=p1.
    // B = ones(4x16)  =>  D[m][n] = sum_k A[m][k] = p0_m + p1_m + p0_{m+16} + p1_{m+16}.
    // D VGPR r: lanes 0-15 hold rowsum s_r, lanes 16-31 hold s_{r+8}.
    v2f a; a.x = p0;   a.y = p1;
    v2f b; b.x = 1.0f; b.y = 1.0f;
    v8f cacc = {};
    v8f d = __builtin_amdgcn_wmma_f32_16x16x4_f32(
        /*neg_a=*/false, a, /*neg_b=*/false, b,
        /*c_mod=*/(short)0, cacc, /*reuse_a=*/false, /*reuse_b=*/false);

    float t = d[0] + d[1] + d[2] + d[3] + d[4] + d[5] + d[6] + d[7];
    // lane 0 holds s_0+...+s_7, lane 16 holds s_8+...+s_15
    float wt = __shfl(t, 0, 32) + __shfl(t, 16, 32);

    __shared__ float wsum[TPB / 32];
    if ((tid & 31) == 0) wsum[tid >> 5] = wt;
    __syncthreads();
    if (tid == 0) {
        float s = 0.0f;
#pragma unroll
        for (int w = 0; w < TPB / 32; ++w) s += wsum[w];
        blockSums[blockIdx.x] = s;
    }
}

__global__ void __launch_bounds__(256) final_reduce(
    const float* __restrict__ blockSums, int n, float* __restrict__ out)
{
    __shared__ float sh[256];
    float s = 0.0f;
    for (int i = threadIdx.x; i < n; i += 256) s += blockSums[i];
    sh[threadIdx.x] = s;
    __syncthreads();
    for (int off = 128; off > 0; off >>= 1) {
        if (threadIdx.x < off) sh[threadIdx.x] += sh[threadIdx.x + off];
        __syncthreads();
    }
    if (threadIdx.x == 0) out[0] = sh[0];
}

extern "C" void kernel_launch(void* const* d_in, const int* in_sizes, int n_in,
                              void* d_out, int out_size, void* d_ws, size_t ws_size,
                              hipStream_t stream) {
    const float* nod = (const float*)d_in[0];   // nodal_values (nNodes, 2)
    const float* xy  = (const float*)d_in[1];   // coords       (nNodes, 2)
    const int*   el  = (const int*)  d_in[2];   // elements     (nElem, 3)
    const float* Nm  = (const float*)d_in[3];   // N            (3, 3)
    const float* dNm = (const float*)d_in[4];   // dN           (3, 3, 2)
    const float* qw  = (const float*)d_in[5];   // quad_weights (3,)

    const int nElem   = in_sizes[2] / 3;
    const int nBlocks = (nElem + TPB * EPT - 1) / (TPB * EPT);
    float* bsums = (float*)d_ws;

    fea_energy_partials<<<nBlocks, TPB, 0, stream>>>(nod, xy, el, Nm, dNm, qw,
                                                     nElem, bsums);
    final_reduce<<<1, 256, 0, stream>>>(bsums, nBlocks, (float*)d_out);
}